// GSSupervised_11158325035270
// MI455X (gfx1250) — compile-verified
//
#include <hip/hip_runtime.h>
#include <hip/hip_bf16.h>

#define NN     1000000
#define DEG    64
#define BATCH  4096
#define NS     25
#define EMB    64
#define HID    128
#define ROWS1  (BATCH * NS)   // 102400

typedef float v2f __attribute__((ext_vector_type(2)));
typedef float v8f __attribute__((ext_vector_type(8)));

// ---------------------------------------------------------------------------
// Generic fp32 WMMA GEMM:  C = act( A[MxK] @ W[NxK]^T (+ bias) )
// One 16x16 output tile per wave, K stepped by 4 via v_wmma_f32_16x16x4_f32.
// meanGroup > 0: instead of storing rows, accumulate val*meanScale into
//                C[row/meanGroup][colOff+n] with grouped atomics (fused mean).
// ---------------------------------------------------------------------------
__global__ void gemm_wmma_f32(const float* __restrict__ A,
                              const float* __restrict__ W,
                              const float* __restrict__ bias,
                              float* __restrict__ C,
                              int M, int N, int K, int ldC, int colOff,
                              int doRelu, int meanGroup, float meanScale)
{
    const int lane  = threadIdx.x & 31;
    const int wave  = threadIdx.x >> 5;
    const int tilesN = N >> 4;
    const int tile  = blockIdx.x * (blockDim.x >> 5) + wave;
    if (tile >= (M >> 4) * tilesN) return;          // wave-uniform: EXEC stays all-1
    const int mt   = tile / tilesN;
    const int nt   = tile - mt * tilesN;
    const int half = lane >> 4;                     // 0: K+{0,1}   1: K+{2,3}
    const int l15  = lane & 15;

    // A frag: row M = mt*16+l15, two consecutive K values per lane
    const float* Ap = A + (size_t)(mt * 16 + l15) * K + half * 2;
    // B frag: B[k][n] = W[n*K + k]  (W is [N x K] row-major)
    const float* Wp = W + (size_t)(nt * 16 + l15) * K + half * 2;

    v8f acc = {0.f, 0.f, 0.f, 0.f, 0.f, 0.f, 0.f, 0.f};
    for (int k = 0; k < K; k += 4) {
        v2f a = *(const v2f*)(Ap + k);
        v2f b = *(const v2f*)(Wp + k);
        acc = __builtin_amdgcn_wmma_f32_16x16x4_f32(false, a, false, b,
                                                    (short)0, acc, false, false);
    }

    const int n  = nt * 16 + l15;
    const float bs = bias ? bias[n] : 0.f;
    float vals[8];
#pragma unroll
    for (int v = 0; v < 8; ++v) {
        float x = acc[v] + bs;
        if (doRelu) x = fmaxf(x, 0.f);
        vals[v] = x;
    }
    const int rowBase = mt * 16 + half * 8;         // VGPR v holds row rowBase+v

    if (meanGroup > 0) {
        // fused mean over meanGroup consecutive rows: group per-lane, then atomic
        int v = 0;
        while (v < 8) {
            int b = (rowBase + v) / meanGroup;
            float s = 0.f;
            while (v < 8 && (rowBase + v) / meanGroup == b) { s += vals[v]; ++v; }
            atomicAdd(C + (size_t)b * ldC + colOff + n, s * meanScale);
        }
    } else {
#pragma unroll
        for (int v = 0; v < 8; ++v)
            C[(size_t)(rowBase + v) * ldC + colOff + n] = vals[v];
    }
}

// ---------------------------------------------------------------------------
// Constants: e_dummy = emb[NN] @ prepW^T + prepB ;  c0 = relu(e_dummy @ w1x^T)
// ---------------------------------------------------------------------------
__global__ void prep_const_kernel(const float* __restrict__ emb,
                                  const float* __restrict__ prepW,
                                  const float* __restrict__ prepB,
                                  const float* __restrict__ w1x,
                                  float* __restrict__ c0)
{
    __shared__ float ed[EMB];
    const int t = threadIdx.x;                      // 128 threads
    if (t < EMB) {
        const float* e = emb + (size_t)NN * EMB;
        float s = prepB[t];
        for (int k = 0; k < EMB; ++k) s += e[k] * prepW[t * EMB + k];
        ed[t] = s;
    }
    __syncthreads();
    float s = 0.f;
    for (int j = 0; j < EMB; ++j) s += ed[j] * w1x[t * EMB + j];
    c0[t] = fmaxf(s, 0.f);
}

// ---------------------------------------------------------------------------
// One wave per hop-1 row i = b*NS + s:
//   emb1[i]   = embedding[adj[ids[b]][perm1[s]]]
//   meanE2[i] = mean_j embedding[adj[id1][perm2[j]]]     (mean BEFORE prep!)
// ---------------------------------------------------------------------------
__global__ void gather_kernel(const float* __restrict__ emb,
                              const int* __restrict__ adj,
                              const int* __restrict__ ids,
                              const int* __restrict__ perm1,
                              const int* __restrict__ perm2,
                              float* __restrict__ emb1,
                              float* __restrict__ meanE2)
{
    const int lane = threadIdx.x & 31;
    const int wave = threadIdx.x >> 5;
    const int i = blockIdx.x * (blockDim.x >> 5) + wave;
    if (i >= ROWS1) return;
    const int b = i / NS;
    const int s = i - b * NS;

    const int id1 = adj[(size_t)ids[b] * DEG + perm1[s]];
    const float* e1 = emb + (size_t)id1 * EMB;
    emb1[(size_t)i * EMB + lane]      = e1[lane];
    emb1[(size_t)i * EMB + 32 + lane] = e1[32 + lane];

    const int* arow = adj + (size_t)id1 * DEG;
    float s0 = 0.f, s1 = 0.f;
    for (int j = 0; j < NS; ++j) {
        const float* e2 = emb + (size_t)arow[perm2[j]] * EMB;
        s0 += e2[lane];
        s1 += e2[32 + lane];
    }
    meanE2[(size_t)i * EMB + lane]      = s0 * (1.f / NS);
    meanE2[(size_t)i * EMB + 32 + lane] = s1 * (1.f / NS);
}

// mean of emb1 rows per batch element -> meanE1 [BATCH x 64]
__global__ void mean_emb1_kernel(const float* __restrict__ emb1,
                                 float* __restrict__ meanE1)
{
    const int lane = threadIdx.x & 31;
    const int wave = threadIdx.x >> 5;
    const int b = blockIdx.x * (blockDim.x >> 5) + wave;
    if (b >= BATCH) return;
    const float* base = emb1 + (size_t)b * NS * EMB;
    float s0 = 0.f, s1 = 0.f;
    for (int j = 0; j < NS; ++j) {
        s0 += base[j * EMB + lane];
        s1 += base[j * EMB + 32 + lane];
    }
    meanE1[(size_t)b * EMB + lane]      = s0 * (1.f / NS);
    meanE1[(size_t)b * EMB + 32 + lane] = s1 * (1.f / NS);
}

// h0[:, :128] = c0 (broadcast, already relu'd); zero mh1 accumulator
__global__ void fill_kernel(float* __restrict__ h0,
                            const float* __restrict__ c0,
                            float* __restrict__ mh1)
{
    const int idx = blockIdx.x * blockDim.x + threadIdx.x;   // BATCH*128 threads
    const int b = idx >> 7;
    const int c = idx & 127;
    h0[(size_t)b * 256 + c] = c0[c];
    mh1[(size_t)idx * 2]     = 0.f;
    mh1[(size_t)idx * 2 + 1] = 0.f;
}

// row-wise L2 normalize + fc:  out[b] = dot(h, fcW)/max(||h||,1e-12) + fcB
__global__ void final_kernel(const float* __restrict__ hbuf,
                             const float* __restrict__ fcW,
                             const float* __restrict__ fcB,
                             float* __restrict__ out)
{
    const int lane = threadIdx.x & 31;
    const int wave = threadIdx.x >> 5;
    const int b = blockIdx.x * (blockDim.x >> 5) + wave;
    if (b >= BATCH) return;
    const float* h = hbuf + (size_t)b * 256;
    float ss = 0.f, dp = 0.f;
    for (int c = lane; c < 256; c += 32) {
        float x = h[c];
        ss += x * x;
        dp += x * fcW[c];
    }
    for (int off = 16; off > 0; off >>= 1) {
        ss += __shfl_xor(ss, off, 32);
        dp += __shfl_xor(dp, off, 32);
    }
    if (lane == 0) out[b] = dp / fmaxf(sqrtf(ss), 1e-12f) + fcB[0];
}

// ---------------------------------------------------------------------------
extern "C" void kernel_launch(void* const* d_in, const int* in_sizes, int n_in,
                              void* d_out, int out_size, void* d_ws, size_t ws_size,
                              hipStream_t stream)
{
    const float* emb   = (const float*)d_in[0];
    const float* prepW = (const float*)d_in[1];
    const float* prepB = (const float*)d_in[2];
    const float* w1x   = (const float*)d_in[3];
    const float* w1n   = (const float*)d_in[4];
    const float* w2x   = (const float*)d_in[5];
    const float* w2n   = (const float*)d_in[6];
    const float* fcW   = (const float*)d_in[7];
    const float* fcB   = (const float*)d_in[8];
    const int*   ids   = (const int*)d_in[9];
    const int*   adj   = (const int*)d_in[10];
    const int*   perm1 = (const int*)d_in[11];
    const int*   perm2 = (const int*)d_in[12];
    float* out = (float*)d_out;

    // workspace layout (floats); all offsets 16B-aligned
    float* ws = (float*)d_ws;
    const size_t R1E = (size_t)ROWS1 * EMB;         // 6,553,600
    float* emb1   = ws;                              // [ROWS1 x 64]
    float* meanE2 = emb1 + R1E;                      // [ROWS1 x 64]
    float* feats1 = meanE2 + R1E;                    // [ROWS1 x 64]
    float* m2     = emb1;                            // reuse emb1 after GEMM1
    float* mh1    = feats1 + R1E;                    // [BATCH x 256]
    float* meanE1 = mh1 + (size_t)BATCH * 256;       // [BATCH x 64]
    float* m1     = meanE1 + (size_t)BATCH * EMB;    // [BATCH x 64]
    float* h0     = m1 + (size_t)BATCH * EMB;        // [BATCH x 256]
    float* hbuf   = h0 + (size_t)BATCH * 256;        // [BATCH x 256]
    float* c0     = hbuf + (size_t)BATCH * 256;      // [128]

    const int TPB = 256, WPB = 8;
    auto gblocks = [](int M, int N) { return ((M >> 4) * (N >> 4) + 7) / 8; };

    // constants + gathers + small means + buffer init
    prep_const_kernel<<<1, 128, 0, stream>>>(emb, prepW, prepB, w1x, c0);
    gather_kernel<<<ROWS1 / WPB, TPB, 0, stream>>>(emb, adj, ids, perm1, perm2,
                                                   emb1, meanE2);
    mean_emb1_kernel<<<BATCH / WPB, TPB, 0, stream>>>(emb1, meanE1);
    fill_kernel<<<BATCH * 128 / TPB, TPB, 0, stream>>>(h0, c0, mh1);

    // prep GEMMs (with bias)
    gemm_wmma_f32<<<gblocks(ROWS1, 64), TPB, 0, stream>>>(
        emb1, prepW, prepB, feats1, ROWS1, 64, 64, 64, 0, 0, 0, 0.f);
    gemm_wmma_f32<<<gblocks(ROWS1, 64), TPB, 0, stream>>>(
        meanE2, prepW, prepB, m2, ROWS1, 64, 64, 64, 0, 0, 0, 0.f);
    gemm_wmma_f32<<<gblocks(BATCH, 64), TPB, 0, stream>>>(
        meanE1, prepW, prepB, m1, BATCH, 64, 64, 64, 0, 0, 0, 0.f);

    // layer 1 for h1 with fused relu + mean over NS -> mh1 [BATCH x 256]
    gemm_wmma_f32<<<gblocks(ROWS1, HID), TPB, 0, stream>>>(
        feats1, w1x, nullptr, mh1, ROWS1, HID, 64, 256, 0, 1, NS, 1.f / NS);
    gemm_wmma_f32<<<gblocks(ROWS1, HID), TPB, 0, stream>>>(
        m2, w1n, nullptr, mh1, ROWS1, HID, 64, 256, 128, 1, NS, 1.f / NS);

    // layer 1 for h0 (right half; left half is broadcast c0)
    gemm_wmma_f32<<<gblocks(BATCH, HID), TPB, 0, stream>>>(
        m1, w1n, nullptr, h0, BATCH, HID, 64, 256, 128, 1, 0, 0.f);

    // layer 2: h = [h0 @ w2x^T | mh1 @ w2n^T]
    gemm_wmma_f32<<<gblocks(BATCH, HID), TPB, 0, stream>>>(
        h0, w2x, nullptr, hbuf, BATCH, HID, 256, 256, 0, 0, 0, 0.f);
    gemm_wmma_f32<<<gblocks(BATCH, HID), TPB, 0, stream>>>(
        mh1, w2n, nullptr, hbuf, BATCH, HID, 256, 256, 128, 0, 0, 0.f);

    // normalize + fc
    final_kernel<<<BATCH / WPB, TPB, 0, stream>>>(hbuf, fcW, fcB, out);
}